// activation_moduleA_48112223650431
// MI455X (gfx1250) — compile-verified
//
#include <hip/hip_runtime.h>

// N = 33554432, SIZE = N/2 = 16777216
// out[0:SIZE]     = p = x[0:SIZE]
// out[SIZE:N]     = weight * selu(p) + q,  q = x[SIZE:N]
//
// Memory-bound streaming kernel: 320 MB total traffic, ~14us floor at 23.3 TB/s.
// Strategy: b128 vector accesses + non-temporal hints (footprint > 192MB L2).

typedef float v4f __attribute__((ext_vector_type(4)));

#define SELU_SCALE 1.0507009873554804934193349852946f
#define SELU_ALPHA 1.6732632423543772848170429916717f

__device__ __forceinline__ float selu_one(float v) {
    // scale * (v > 0 ? v : alpha*(exp(v)-1))
    float neg = (SELU_SCALE * SELU_ALPHA) * (__expf(v) - 1.0f);
    float pos = SELU_SCALE * v;
    return v > 0.0f ? pos : neg;
}

__global__ __launch_bounds__(256) void selu_residual_kernel(
    const v4f* __restrict__ p4,
    const v4f* __restrict__ q4,
    const v4f* __restrict__ w4,
    v4f* __restrict__ out_p4,
    v4f* __restrict__ out_q4,
    int n4)
{
    int i = blockIdx.x * blockDim.x + threadIdx.x;
    if (i >= n4) return;

    // Streaming (non-temporal) 128-bit loads: data touched exactly once.
    v4f p = __builtin_nontemporal_load(&p4[i]);
    v4f q = __builtin_nontemporal_load(&q4[i]);
    v4f w = __builtin_nontemporal_load(&w4[i]);

    v4f r;
    r.x = __builtin_fmaf(w.x, selu_one(p.x), q.x);
    r.y = __builtin_fmaf(w.y, selu_one(p.y), q.y);
    r.z = __builtin_fmaf(w.z, selu_one(p.z), q.z);
    r.w = __builtin_fmaf(w.w, selu_one(p.w), q.w);

    // First half of output is a straight copy of p (already in registers).
    __builtin_nontemporal_store(p, &out_p4[i]);
    __builtin_nontemporal_store(r, &out_q4[i]);
}

extern "C" void kernel_launch(void* const* d_in, const int* in_sizes, int n_in,
                              void* d_out, int out_size, void* d_ws, size_t ws_size,
                              hipStream_t stream) {
    const float* x      = (const float*)d_in[0];   // N floats
    const float* weight = (const float*)d_in[1];   // SIZE floats
    float*       out    = (float*)d_out;           // N floats

    const int N    = in_sizes[0];      // 33554432
    const int SIZE = N / 2;            // 16777216
    const int n4   = SIZE / 4;         // 4194304 float4 elements

    const v4f* p4     = (const v4f*)x;
    const v4f* q4     = (const v4f*)(x + SIZE);
    const v4f* w4     = (const v4f*)weight;
    v4f*       out_p4 = (v4f*)out;
    v4f*       out_q4 = (v4f*)(out + SIZE);

    const int threads = 256;
    const int blocks  = (n4 + threads - 1) / threads;

    selu_residual_kernel<<<blocks, threads, 0, stream>>>(p4, q4, w4, out_p4, out_q4, n4);
}